// FeatureEncodingLayer_30374008718005
// MI455X (gfx1250) — compile-verified
//
#include <hip/hip_runtime.h>

typedef float v2f __attribute__((ext_vector_type(2)));
typedef float v8f __attribute__((ext_vector_type(8)));

#define BATCH 1024
#define UNITS 4096
#define INDIM 256

// One wave computes a 16(batch) x 16(units) tile of theta via V_WMMA_F32_16X16X4_F32,
// converts to (c,s,o) with one __sincosf + half-angle identities, stages in LDS,
// then recombines even/odd unit pairs into the 4x4 complex Kronecker blocks and
// streams them out with non-temporal stores.
__global__ __launch_bounds__(256) void fel_kernel(const float* __restrict__ X,
                                                  const float* __restrict__ W,
                                                  const float* __restrict__ bias,
                                                  v2f* __restrict__ out)
{
    // per-wave staging: [wave][{c,s,o}][m][n]
    __shared__ float lds[8][3][16][16];

    const int lane = threadIdx.x & 31;
    const int wave = threadIdx.x >> 5;
    const int wgid = blockIdx.x * 8 + wave;     // global wave id, 16384 total
    const int nTile = wgid & 255;               // 256 unit-tiles
    const int mTile = wgid >> 8;                // 64 batch-tiles
    const int m0 = mTile * 16;
    const int n0 = nTile * 16;

    const int lmod = lane & 15;                 // row (A) / col (B,C,D)
    const int lhi  = lane >> 4;                 // K-half selector

    // A: lane holds X[m0+lmod][k + 2*lhi + {0,1}]  (16x4 f32 fragment layout)
    // B: lane holds W[n0+lmod][k + 2*lhi + {0,1}]  (4x16 fragment, B = W^T slice)
    const float* aPtr = X + (size_t)(m0 + lmod) * INDIM + 2 * lhi;
    const float* bPtr = W + (size_t)(n0 + lmod) * INDIM + 2 * lhi;

    v8f acc = {};
    #pragma unroll 4
    for (int kk = 0; kk < INDIM; kk += 4) {
        v2f a = *(const v2f*)(aPtr + kk);
        v2f b = *(const v2f*)(bPtr + kk);
        // D = A(16x4) * B(4x16) + C, exact fp32
        acc = __builtin_amdgcn_wmma_f32_16x16x4_f32(
            /*neg_a=*/false, a, /*neg_b=*/false, b,
            /*c_mod=*/(short)0, acc, /*reuse_a=*/false, /*reuse_b=*/false);
    }

    const float bv = bias[n0 + lmod];

    // C/D layout: vgpr v, lane L -> M = v + 8*(L/16), N = L%16
    #pragma unroll
    for (int v = 0; v < 8; ++v) {
        float theta = acc[v] + bv;
        float sn, cs;
        __sincosf(theta, &sn, &cs);
        // cos^2(t/2) = (1+cos t)/2 ; sin^2(t/2) = (1-cos t)/2 ; o = sin(t)/2
        const int m = v + 8 * lhi;
        lds[wave][0][m][lmod] = 0.5f * (1.0f + cs);
        lds[wave][1][m][lmod] = 0.5f * (1.0f - cs);
        lds[wave][2][m][lmod] = 0.5f * sn;
    }
    __syncthreads();

    // Output element out[bm][i][j][p], complex64:
    //   R(u) = [[c, i*o], [-i*o, s]];  A = R(2p), B = R(2p+1)
    //   kron[i][j] = A[i>>1][j>>1] * B[i&1][j&1]
    const int pl = lane & 7;        // pair index within tile (8 pairs)
    const int j  = lane >> 3;       // output column 0..3 (fixed per lane)
    const int jA = j >> 1, jB = j & 1;
    const int nEven = 2 * pl, nOdd = nEven + 1;
    const long pGlob = (long)nTile * 8 + pl;

    #pragma unroll
    for (int m = 0; m < 16; ++m) {
        const float cA = lds[wave][0][m][nEven];
        const float sA = lds[wave][1][m][nEven];
        const float oA = lds[wave][2][m][nEven];
        const float cB = lds[wave][0][m][nOdd];
        const float sB = lds[wave][1][m][nOdd];
        const float oB = lds[wave][2][m][nOdd];

        // R rows at fixed column jA / jB:  row0: (c,0) | (0,o) ; row1: (0,-o) | (s,0)
        v2f A0, A1, B0, B1;
        A0.x = (jA == 0) ? cA : 0.0f;   A0.y = (jA == 0) ? 0.0f : oA;
        A1.x = (jA == 0) ? 0.0f : sA;   A1.y = (jA == 0) ? -oA : 0.0f;
        B0.x = (jB == 0) ? cB : 0.0f;   B0.y = (jB == 0) ? 0.0f : oB;
        B1.x = (jB == 0) ? 0.0f : sB;   B1.y = (jB == 0) ? -oB : 0.0f;

        const long bm = m0 + m;
        #pragma unroll
        for (int i = 0; i < 4; ++i) {
            v2f Ae = (i < 2) ? A0 : A1;
            v2f Be = (i & 1) ? B1 : B0;
            v2f r;
            r.x = Ae.x * Be.x - Ae.y * Be.y;
            r.y = Ae.x * Be.y + Ae.y * Be.x;
            const long idx = ((bm * 4 + i) * 4 + j) * 2048 + pGlob;
            // 268 MB streaming output, never re-read: non-temporal store
            __builtin_nontemporal_store(r, out + idx);
        }
    }
}

extern "C" void kernel_launch(void* const* d_in, const int* in_sizes, int n_in,
                              void* d_out, int out_size, void* d_ws, size_t ws_size,
                              hipStream_t stream) {
    const float* X    = (const float*)d_in[0];   // (1024, 256) f32
    const float* W    = (const float*)d_in[1];   // (4096, 256) f32
    const float* bias = (const float*)d_in[2];   // (4096,)     f32
    v2f* out = (v2f*)d_out;                      // (1024,4,4,2048) complex64 -> float2

    // 64 m-tiles * 256 n-tiles = 16384 waves ; 8 waves/block -> 2048 blocks
    fel_kernel<<<dim3(2048), dim3(256), 0, stream>>>(X, W, bias, out);
}